// MPNN_GNN2D_L_K_86878598464027
// MI455X (gfx1250) — compile-verified
//
#include <hip/hip_runtime.h>

typedef __attribute__((ext_vector_type(2))) float v2f;
typedef __attribute__((ext_vector_type(8))) float v8f;

namespace {

constexpr int kBR = 512;   // B*R
constexpr int kL  = 64;
constexpr int kK  = 16;
constexpr int kU  = 4;
constexpr int kE  = 64;
constexpr int kC2 = 8;     // 2U
constexpr float kEps = 1e-5f;
constexpr float kInvN = 1.0f / (512.0f * 64.0f * 16.0f);  // BN count

// workspace layout (float offsets)
constexpr int OFF_HML = 0;                         // [512][8][64]
constexpr int OFF_HMK = OFF_HML + kBR * kC2 * kL;  // [512][8][16]
constexpr int OFF_ST0 = OFF_HMK + kBR * kC2 * kK;  // sum[64] sumsq[64]
constexpr int OFF_ST1 = OFF_ST0 + 128;
constexpr int OFF_SS0 = OFF_ST1 + 128;             // scale[64] shift[64]
constexpr int OFF_SS1 = OFF_SS0 + 128;

__device__ inline v8f wmma_f32(v2f a, v2f b, v8f c) {
  // D = A(16x4,f32) * B(4x16,f32) + C(16x16,f32)
  return __builtin_amdgcn_wmma_f32_16x16x4_f32(false, a, false, b, (short)0, c,
                                               false, false);
}

// ---------------------------------------------------------------------------
// shared helpers (operate on per-workgroup LDS arrays, blockDim.x == 256)
// ---------------------------------------------------------------------------
__device__ inline void load_hm(int br, const float* ws,
                               float (&sHmL)[kC2 * kL],
                               float (&sHmK)[kC2 * kK]) {
  const float* HmL = ws + OFF_HML + br * kC2 * kL;
  const float* HmK = ws + OFF_HMK + br * kC2 * kK;
  for (int i = threadIdx.x; i < kC2 * kL; i += blockDim.x) sHmL[i] = HmL[i];
  for (int i = threadIdx.x; i < kC2 * kK; i += blockDim.x) sHmK[i] = HmK[i];
}

// layer0 is separable: pre0[e,l,k] = X0[e,l] + Y0[e,k]
__device__ inline void compute_xy0(const float* __restrict__ P4_0,
                                   const float (&sHmL)[kC2 * kL],
                                   const float (&sHmK)[kC2 * kK],
                                   float (&sX0)[kE][kL],
                                   float (&sY0)[kE][kK]) {
  for (int i = threadIdx.x; i < kE * kL; i += blockDim.x) {
    int e = i >> 6, l = i & 63;
    float s = 0.f;
#pragma unroll
    for (int c = 0; c < kC2; ++c) s += P4_0[e * kC2 + c] * sHmL[c * kL + l];
    sX0[e][l] = 0.1f * s;
  }
  for (int i = threadIdx.x; i < kE * kK; i += blockDim.x) {
    int e = i >> 4, k = i & 15;
    float s = 0.f;
#pragma unroll
    for (int c = 0; c < kC2; ++c) s += P4_0[e * kC2 + c] * sHmK[c * kK + k];
    sY0[e][k] = 0.1f * s;
  }
}

// means of A0 and the l/k-broadcast bias terms of layer 1
__device__ inline void build_layer1_bias(const float* __restrict__ P2_1,
                                         const float* __restrict__ P4_1,
                                         const float (&sX0)[kE][kL],
                                         const float (&sY0)[kE][kK],
                                         const float (&sScale0)[kE],
                                         const float (&sShift0)[kE],
                                         const float (&sHmL)[kC2 * kL],
                                         const float (&sHmK)[kC2 * kK],
                                         float (&sMK0)[kE][kL],
                                         float (&sML0)[kE][kK],
                                         float (&sB1L)[kE][kL],
                                         float (&sB1K)[kE][kK]) {
  // meanK(A0)[e][l], meanL(A0)[e][k] with BN0 affine folded in
  for (int i = threadIdx.x; i < kE * kL; i += blockDim.x) {
    int e = i >> 6, l = i & 63;
    float x = sX0[e][l], s = 0.f;
#pragma unroll
    for (int k = 0; k < kK; ++k) s += fmaxf(x + sY0[e][k], 0.f);
    sMK0[e][l] = sScale0[e] * s * (1.0f / kK) + sShift0[e];
  }
  for (int i = threadIdx.x; i < kE * kK; i += blockDim.x) {
    int e = i >> 4, k = i & 15;
    float y = sY0[e][k], s = 0.f;
#pragma unroll
    for (int l = 0; l < kL; ++l) s += fmaxf(sX0[e][l] + y, 0.f);
    sML0[e][k] = sScale0[e] * s * (1.0f / kL) + sShift0[e];
  }
  __syncthreads();
  for (int i = threadIdx.x; i < kE * kL; i += blockDim.x) {
    int o = i >> 6, l = i & 63;
    float s = 0.f;
    for (int e = 0; e < kE; ++e) s += P2_1[o * kE + e] * sMK0[e][l];
    float h = 0.f;
#pragma unroll
    for (int c = 0; c < kC2; ++c) h += P4_1[o * kC2 + c] * sHmL[c * kL + l];
    sB1L[o][l] = 2.f * s + 0.1f * h;
  }
  for (int i = threadIdx.x; i < kE * kK; i += blockDim.x) {
    int o = i >> 4, k = i & 15;
    float s = 0.f;
    for (int e = 0; e < kE; ++e) s += P2_1[o * kE + e] * sML0[e][k];
    float h = 0.f;
#pragma unroll
    for (int c = 0; c < kC2; ++c) h += P4_1[o * kC2 + c] * sHmK[c * kK + k];
    sB1K[o][k] = 2.f * s + 0.1f * h;
  }
}

// WMMA: acc[o_tile 16 x (k=0..15) ] = sum_e P1_1[o,e] * A0[e, l, k]
// A0 reconstructed on the fly from X0/Y0 + BN0 affine.
__device__ inline v8f compute_p1A0_tile(int ot, int l, int half, int li,
                                        const float (&sP1)[kE][68],
                                        const float (&sX0)[kE][kL],
                                        const float (&sY0)[kE][kK],
                                        const float (&sScale0)[kE],
                                        const float (&sShift0)[kE]) {
  v8f acc = {0.f, 0.f, 0.f, 0.f, 0.f, 0.f, 0.f, 0.f};
  const int orow = ot * 16 + li;
#pragma unroll
  for (int ke = 0; ke < 16; ++ke) {
    const int e0 = ke * 4 + 2 * half;
    v2f a, b;
    a.x = sP1[orow][e0];
    a.y = sP1[orow][e0 + 1];
    b.x = sScale0[e0] * fmaxf(sX0[e0][l] + sY0[e0][li], 0.f) + sShift0[e0];
    b.y = sScale0[e0 + 1] * fmaxf(sX0[e0 + 1][l] + sY0[e0 + 1][li], 0.f) +
          sShift0[e0 + 1];
    acc = wmma_f32(a, b, acc);
  }
  return acc;
}

}  // namespace

// ---------------------------------------------------------------------------
// K0: zero BN stats accumulators
// ---------------------------------------------------------------------------
__global__ void k_zero_stats(float* __restrict__ ws) {
  int t = threadIdx.x;
  if (t < 256) ws[OFF_ST0 + t] = 0.f;  // ST0(128) + ST1(128) contiguous
}

// ---------------------------------------------------------------------------
// K1: per-image means of Hup over K and over L
// ---------------------------------------------------------------------------
__global__ __launch_bounds__(256) void k_hmeans(const float* __restrict__ Hr,
                                                const float* __restrict__ Hi,
                                                float* __restrict__ ws) {
  const int br = blockIdx.x;
  float* HmL = ws + OFF_HML + br * kC2 * kL;
  float* HmK = ws + OFF_HMK + br * kC2 * kK;
  const float* hr = Hr + (size_t)br * kL * kK * kU;
  const float* hi = Hi + (size_t)br * kL * kK * kU;
  for (int idx = threadIdx.x; idx < kC2 * kL; idx += blockDim.x) {
    int c = idx / kL, l = idx % kL;
    const float* src = (c < kU) ? hr : hi;
    int u = (c < kU) ? c : c - kU;
    float s = 0.f;
#pragma unroll
    for (int k = 0; k < kK; ++k) s += src[(l * kK + k) * kU + u];
    HmL[idx] = s * (1.0f / kK);
  }
  for (int idx = threadIdx.x; idx < kC2 * kK; idx += blockDim.x) {
    int c = idx / kK, k = idx % kK;
    const float* src = (c < kU) ? hr : hi;
    int u = (c < kU) ? c : c - kU;
    float s = 0.f;
    for (int l = 0; l < kL; ++l) s += src[(l * kK + k) * kU + u];
    HmK[idx] = s * (1.0f / kL);
  }
}

// ---------------------------------------------------------------------------
// K2: BN0 statistics (sum / sumsq of relu(X0+Y0) per channel)
// ---------------------------------------------------------------------------
__global__ __launch_bounds__(256) void k_bn0_stats(
    const float* __restrict__ P4_0, float* __restrict__ ws) {
  __shared__ float sHmL[kC2 * kL];
  __shared__ float sHmK[kC2 * kK];
  __shared__ float sX0[kE][kL];
  __shared__ float sY0[kE][kK];
  __shared__ float sPS[kE][4];
  __shared__ float sPQ[kE][4];
  const int br = blockIdx.x;
  load_hm(br, ws, sHmL, sHmK);
  __syncthreads();
  compute_xy0(P4_0, sHmL, sHmK, sX0, sY0);
  __syncthreads();
  const int t = threadIdx.x;
  const int e = t >> 2, q = t & 3;
  float s = 0.f, sq = 0.f;
  for (int l = q * 16; l < q * 16 + 16; ++l) {
    float x = sX0[e][l];
#pragma unroll
    for (int k = 0; k < kK; ++k) {
      float v = fmaxf(x + sY0[e][k], 0.f);
      s += v;
      sq += v * v;
    }
  }
  sPS[e][q] = s;
  sPQ[e][q] = sq;
  __syncthreads();
  if (t < kE) {
    atomicAdd(&ws[OFF_ST0 + t], sPS[t][0] + sPS[t][1] + sPS[t][2] + sPS[t][3]);
    atomicAdd(&ws[OFF_ST0 + 64 + t],
              sPQ[t][0] + sPQ[t][1] + sPQ[t][2] + sPQ[t][3]);
  }
}

// ---------------------------------------------------------------------------
// K3/K5: BN finalize -> per-channel scale/shift
// ---------------------------------------------------------------------------
__global__ void k_bn_finalize(const float* __restrict__ g,
                              const float* __restrict__ b,
                              float* __restrict__ ws, int statOff, int ssOff) {
  int t = threadIdx.x;
  if (t < kE) {
    float mu = ws[statOff + t] * kInvN;
    float var = ws[statOff + 64 + t] * kInvN - mu * mu;
    float sc = g[t] * rsqrtf(var + kEps);
    ws[ssOff + t] = sc;
    ws[ssOff + 64 + t] = b[t] - mu * sc;
  }
}

// ---------------------------------------------------------------------------
// K4: layer-1 forward (WMMA) -> BN1 statistics only (no spill of A1 to HBM)
// ---------------------------------------------------------------------------
__global__ __launch_bounds__(256) void k_bn1_stats(
    const float* __restrict__ P4_0, const float* __restrict__ P1_1,
    const float* __restrict__ P2_1, const float* __restrict__ P4_1,
    float* __restrict__ ws) {
  __shared__ float sHmL[kC2 * kL];
  __shared__ float sHmK[kC2 * kK];
  __shared__ float sX0[kE][kL];
  __shared__ float sY0[kE][kK];
  __shared__ float sScale0[kE], sShift0[kE];
  __shared__ float sMK0[kE][kL];
  __shared__ float sML0[kE][kK];
  __shared__ float sB1L[kE][kL];
  __shared__ float sB1K[kE][kK];
  __shared__ float sP1[kE][68];  // pad 68: conflict-free column reads
  __shared__ float sSum[kE], sSq[kE];

  const int br = blockIdx.x;
  load_hm(br, ws, sHmL, sHmK);
  if (threadIdx.x < kE) {
    sScale0[threadIdx.x] = ws[OFF_SS0 + threadIdx.x];
    sShift0[threadIdx.x] = ws[OFF_SS0 + 64 + threadIdx.x];
    sSum[threadIdx.x] = 0.f;
    sSq[threadIdx.x] = 0.f;
  }
  for (int i = threadIdx.x; i < kE * kE; i += blockDim.x)
    sP1[i >> 6][i & 63] = P1_1[i];
  __syncthreads();
  compute_xy0(P4_0, sHmL, sHmK, sX0, sY0);
  __syncthreads();
  build_layer1_bias(P2_1, P4_1, sX0, sY0, sScale0, sShift0, sHmL, sHmK, sMK0,
                    sML0, sB1L, sB1K);
  __syncthreads();

  const int wave = threadIdx.x >> 5;
  const int lane = threadIdx.x & 31;
  const int half = lane >> 4, li = lane & 15;
  for (int tile = wave; tile < 4 * kL; tile += 8) {
    const int ot = tile & 3, l = tile >> 2;
    v8f acc = compute_p1A0_tile(ot, l, half, li, sP1, sX0, sY0, sScale0, sShift0);
#pragma unroll
    for (int r = 0; r < 8; ++r) {
      const int o = ot * 16 + r + 8 * half;
      float v = fmaxf(2.f * acc[r] + sB1L[o][l] + sB1K[o][li], 0.f);
      float s = v, q = v * v;
#pragma unroll
      for (int m = 1; m < 16; m <<= 1) {
        s += __shfl_xor(s, m, 32);
        q += __shfl_xor(q, m, 32);
      }
      if (li == 0) {
        atomicAdd(&sSum[o], s);
        atomicAdd(&sSq[o], q);
      }
    }
  }
  __syncthreads();
  if (threadIdx.x < kE) {
    atomicAdd(&ws[OFF_ST1 + threadIdx.x], sSum[threadIdx.x]);
    atomicAdd(&ws[OFF_ST1 + 64 + threadIdx.x], sSq[threadIdx.x]);
  }
}

// ---------------------------------------------------------------------------
// K6: recompute layer-1 (WMMA), layer-1 means, layer-2 (WMMA), normalize, emit
// ---------------------------------------------------------------------------
__global__ __launch_bounds__(256) void k_final(
    const float* __restrict__ P4_0, const float* __restrict__ P1_1,
    const float* __restrict__ P2_1, const float* __restrict__ P4_1,
    const float* __restrict__ P1_2, const float* __restrict__ P2_2,
    const float* __restrict__ P4_2, const float* __restrict__ ws,
    float* __restrict__ out) {
  __shared__ float sHmL[kC2 * kL];
  __shared__ float sHmK[kC2 * kK];
  __shared__ float sX0[kE][kL];
  __shared__ float sY0[kE][kK];
  __shared__ float sScale0[kE], sShift0[kE];
  __shared__ float sScale1[kE], sShift1[kE];
  __shared__ float sMK0[kE][kL];
  __shared__ float sML0[kE][kK];
  __shared__ float sB1L[kE][kL];
  __shared__ float sB1K[kE][kK];
  __shared__ float sP1[kE][68];
  __shared__ float sP12[16][68];  // rows 8..15 zero-padded
  __shared__ float sMK1[kE][kL];
  __shared__ float sML1[kE][kK];
  __shared__ float sB2L[kC2][kL];
  __shared__ float sB2K[kC2][kK];
  __shared__ float stg[8][16][17];  // per-wave A1-tile relayout staging

  const int br = blockIdx.x;
  load_hm(br, ws, sHmL, sHmK);
  if (threadIdx.x < kE) {
    sScale0[threadIdx.x] = ws[OFF_SS0 + threadIdx.x];
    sShift0[threadIdx.x] = ws[OFF_SS0 + 64 + threadIdx.x];
    sScale1[threadIdx.x] = ws[OFF_SS1 + threadIdx.x];
    sShift1[threadIdx.x] = ws[OFF_SS1 + 64 + threadIdx.x];
  }
  for (int i = threadIdx.x; i < kE * kE; i += blockDim.x)
    sP1[i >> 6][i & 63] = P1_1[i];
  for (int i = threadIdx.x; i < 16 * 64; i += blockDim.x) {
    int r = i >> 6, c = i & 63;
    sP12[r][c] = (r < kC2) ? P1_2[r * kE + c] : 0.f;
  }
  for (int i = threadIdx.x; i < kE * kK; i += blockDim.x)
    sML1[i >> 4][i & 15] = 0.f;
  __syncthreads();
  compute_xy0(P4_0, sHmL, sHmK, sX0, sY0);
  __syncthreads();
  build_layer1_bias(P2_1, P4_1, sX0, sY0, sScale0, sShift0, sHmL, sHmK, sMK0,
                    sML0, sB1L, sB1K);
  __syncthreads();

  const int wave = threadIdx.x >> 5;
  const int lane = threadIdx.x & 31;
  const int half = lane >> 4, li = lane & 15;

  // ---- pass 1: layer-1 means (meanK exact per tile; meanL via LDS atomics)
  for (int tile = wave; tile < 4 * kL; tile += 8) {
    const int ot = tile & 3, l = tile >> 2;
    v8f acc = compute_p1A0_tile(ot, l, half, li, sP1, sX0, sY0, sScale0, sShift0);
#pragma unroll
    for (int r = 0; r < 8; ++r) {
      const int o = ot * 16 + r + 8 * half;
      float v = fmaxf(2.f * acc[r] + sB1L[o][l] + sB1K[o][li], 0.f);
      float a1 = sScale1[o] * v + sShift1[o];
      float s = a1;
#pragma unroll
      for (int m = 1; m < 16; m <<= 1) s += __shfl_xor(s, m, 32);
      if (li == 0) sMK1[o][l] = s * (1.0f / kK);
      atomicAdd(&sML1[o][li], a1 * (1.0f / kL));
    }
  }
  __syncthreads();

  // ---- layer-2 broadcast bias terms
  for (int i = threadIdx.x; i < kC2 * kL; i += blockDim.x) {
    int p = i >> 6, l = i & 63;
    float s = 0.f;
    for (int o = 0; o < kE; ++o) s += P2_2[p * kE + o] * sMK1[o][l];
    float h = 0.f;
#pragma unroll
    for (int c = 0; c < kC2; ++c) h += P4_2[p * kC2 + c] * sHmL[c * kL + l];
    sB2L[p][l] = 2.f * s + 0.1f * h;
  }
  for (int i = threadIdx.x; i < kC2 * kK; i += blockDim.x) {
    int p = i >> 4, k = i & 15;
    float s = 0.f;
    for (int o = 0; o < kE; ++o) s += P2_2[p * kE + o] * sML1[o][k];
    float h = 0.f;
#pragma unroll
    for (int c = 0; c < kC2; ++c) h += P4_2[p * kC2 + c] * sHmK[c * kK + k];
    sB2K[p][k] = 2.f * s + 0.1f * h;
  }
  __syncthreads();

  // ---- pass 2: recompute A1 tiles, feed layer-2 WMMA, emit normalized output
  for (int l = wave; l < kL; l += 8) {
    v8f acc2 = {0.f, 0.f, 0.f, 0.f, 0.f, 0.f, 0.f, 0.f};
#pragma unroll 1
    for (int ot = 0; ot < 4; ++ot) {
      v8f acc1 =
          compute_p1A0_tile(ot, l, half, li, sP1, sX0, sY0, sScale0, sShift0);
#pragma unroll
      for (int r = 0; r < 8; ++r) {
        const int o = ot * 16 + r + 8 * half;
        float v = fmaxf(2.f * acc1[r] + sB1L[o][l] + sB1K[o][li], 0.f);
        stg[wave][r + 8 * half][li] = sScale1[o] * v + sShift1[o];
      }
      // wave-private staging: compiler inserts s_wait_dscnt for RAW
#pragma unroll
      for (int cc = 0; cc < 4; ++cc) {
        const int ob = cc * 4 + 2 * half;
        v2f a2, b2;
        a2.x = sP12[li][ot * 16 + ob];
        a2.y = sP12[li][ot * 16 + ob + 1];
        b2.x = stg[wave][ob][li];
        b2.y = stg[wave][ob + 1][li];
        acc2 = wmma_f32(a2, b2, acc2);
      }
    }
    if (half == 0) {  // rows p=0..7 live in lanes 0..15
      const int k = li;
      float F1[kU], F2[kU], nrm = 0.f;
#pragma unroll
      for (int u = 0; u < kU; ++u) {
        F1[u] = 2.f * acc2[u] + sB2L[u][l] + sB2K[u][k];
        F2[u] = 2.f * acc2[u + 4] + sB2L[u + 4][l] + sB2K[u + 4][k];
        nrm += F1[u] * F1[u] + F2[u] * F2[u];
      }
      const float inv = rsqrtf(nrm);
      const size_t base = ((((size_t)br * kL + l) * kK + k) * kU) * 2;
#pragma unroll
      for (int u = 0; u < kU; ++u) {
        out[base + 2 * u] = F1[u] * inv;      // real
        out[base + 2 * u + 1] = F2[u] * inv;  // imag
      }
    }
  }
}

// ---------------------------------------------------------------------------
extern "C" void kernel_launch(void* const* d_in, const int* in_sizes, int n_in,
                              void* d_out, int out_size, void* d_ws,
                              size_t ws_size, hipStream_t stream) {
  const float* Hr = (const float*)d_in[0];
  const float* Hi = (const float*)d_in[1];
  // d_in[2]=P1_0, d_in[3]=P2_0 are dead (layer-0 input A == 0)
  const float* P4_0 = (const float*)d_in[4];
  const float* g0 = (const float*)d_in[5];
  const float* b0 = (const float*)d_in[6];
  const float* P1_1 = (const float*)d_in[7];
  const float* P2_1 = (const float*)d_in[8];
  const float* P4_1 = (const float*)d_in[9];
  const float* g1 = (const float*)d_in[10];
  const float* b1 = (const float*)d_in[11];
  const float* P1_2 = (const float*)d_in[12];
  const float* P2_2 = (const float*)d_in[13];
  const float* P4_2 = (const float*)d_in[14];
  float* ws = (float*)d_ws;
  float* out = (float*)d_out;

  k_zero_stats<<<1, 256, 0, stream>>>(ws);
  k_hmeans<<<kBR, 256, 0, stream>>>(Hr, Hi, ws);
  k_bn0_stats<<<kBR, 256, 0, stream>>>(P4_0, ws);
  k_bn_finalize<<<1, 64, 0, stream>>>(g0, b0, ws, OFF_ST0, OFF_SS0);
  k_bn1_stats<<<kBR, 256, 0, stream>>>(P4_0, P1_1, P2_1, P4_1, ws);
  k_bn_finalize<<<1, 64, 0, stream>>>(g1, b1, ws, OFF_ST1, OFF_SS1);
  k_final<<<kBR, 256, 0, stream>>>(P4_0, P1_1, P2_1, P4_1, P1_2, P2_2, P4_2, ws,
                                   out);
}